// GNNMaxCut_90134183674022
// MI455X (gfx1250) — compile-verified
//
#include <hip/hip_runtime.h>
#include <hip/hip_bf16.h>
#include <math.h>

#define IN_DIM 128
#define HID 128

typedef __attribute__((ext_vector_type(2))) float v2f;
typedef __attribute__((ext_vector_type(8))) float v8f;

// ---------------------------------------------------------------- degree ----
__global__ __launch_bounds__(256) void k_init_deg(float* __restrict__ deg, int n) {
    int i = blockIdx.x * blockDim.x + threadIdx.x;
    if (i < n) deg[i] = 1.0f;  // self-loop contributes 1 to every node
}

__global__ __launch_bounds__(256) void k_deg_accum(const int* __restrict__ dst,
                                                   float* __restrict__ deg, int e) {
    int i = blockIdx.x * blockDim.x + threadIdx.x;
    if (i < e) atomicAdd(&deg[dst[i]], 1.0f);
}

__global__ __launch_bounds__(256) void k_dinv(float* __restrict__ deg, int n) {
    int i = blockIdx.x * blockDim.x + threadIdx.x;
    if (i < n) deg[i] = rsqrtf(deg[i]);  // deg >= 1 always (self-loop)
}

// ------------------------------------------------- GEMM1: hW = x @ W1 -------
// f32 WMMA 16x16x4. One wave owns two 16x16 output tiles (N-tiles w and w+4).
// A fragment (16x4): lanes 0-15 hold K={k,k+1}, lanes 16-31 hold K={k+2,k+3}.
// B fragment (4x16): same half-wave split over K; row striped across lanes.
// C/D: 8 VGPRs, lanes 0-15 -> M=v, lanes 16-31 -> M=v+8.
__global__ __launch_bounds__(128) void k_gemm1(const float* __restrict__ X,
                                               const float* __restrict__ W,
                                               float* __restrict__ H, int n) {
    const int wave = threadIdx.x >> 5;   // 0..3
    const int lane = threadIdx.x & 31;
    const int half = lane >> 4;          // 0 or 1
    const int l16  = lane & 15;
    const int m0   = blockIdx.x * 16;
    if (m0 >= n) return;

    const float* xrow = X + (size_t)(m0 + l16) * IN_DIM;
    const int col0 = wave * 16 + l16;          // N-tile `wave`
    const int col1 = (wave + 4) * 16 + l16;    // N-tile `wave+4`

    v8f c0 = {};
    v8f c1 = {};
    for (int k = 0; k < IN_DIM; k += 4) {
        const int kk = k + 2 * half;
        v2f a;
        a.x = xrow[kk];
        a.y = xrow[kk + 1];
        v2f b0, b1;
        b0.x = W[(size_t)kk * HID + col0];
        b0.y = W[(size_t)(kk + 1) * HID + col0];
        b1.x = W[(size_t)kk * HID + col1];
        b1.y = W[(size_t)(kk + 1) * HID + col1];
        c0 = __builtin_amdgcn_wmma_f32_16x16x4_f32(false, a, false, b0,
                                                   (short)0, c0, false, false);
        c1 = __builtin_amdgcn_wmma_f32_16x16x4_f32(false, a, false, b1,
                                                   (short)0, c1, false, false);
    }
#pragma unroll
    for (int v = 0; v < 8; ++v) {
        const int row = m0 + v + 8 * half;
        H[(size_t)row * HID + col0] = c0[v];
        H[(size_t)row * HID + col1] = c1[v];
    }
}

// -------------------------------------------- layer-1 aggregation -----------
// acc1[i][j] = b1[j] + dinv[i]^2 * hW[i][j]   (self-loop term fused into init)
__global__ __launch_bounds__(256) void k_init_acc1(const float* __restrict__ hW,
                                                   const float* __restrict__ dinv,
                                                   const float* __restrict__ b1,
                                                   float* __restrict__ acc, int n) {
    int idx = blockIdx.x * blockDim.x + threadIdx.x;
    if (idx < n * HID) {
        int i = idx >> 7;
        int j = idx & (HID - 1);
        float di = dinv[i];
        acc[idx] = b1[j] + di * di * hW[idx];
    }
}

// one block per edge, 128 threads = one feature each; coalesced gather+scatter
__global__ __launch_bounds__(128) void k_edge1(const int* __restrict__ ei,
                                               const float* __restrict__ dinv,
                                               const float* __restrict__ hW,
                                               float* __restrict__ acc, int e) {
    int edge = blockIdx.x;
    if (edge >= e) return;
    int s = ei[edge];
    int d = ei[e + edge];
    float nrm = dinv[s] * dinv[d];
    int j = threadIdx.x;
    atomicAdd(&acc[(size_t)d * HID + j], nrm * hW[(size_t)s * HID + j]);
}

// ------------------------- layer 2: relu-on-read dot + self-loop init -------
__global__ __launch_bounds__(128) void k_layer2(const float* __restrict__ acc1,
                                                const float* __restrict__ dinv,
                                                const float* __restrict__ W2,
                                                const float* __restrict__ b2,
                                                float* __restrict__ hw2,
                                                float* __restrict__ acc2, int n) {
    int i = blockIdx.x * blockDim.x + threadIdx.x;
    if (i >= n) return;
    const float4* row = (const float4*)(acc1 + (size_t)i * HID);
    float s0 = 0.f, s1 = 0.f;
#pragma unroll
    for (int q = 0; q < HID / 4; ++q) {
        float4 h4 = row[q];
        float h;
        h = fmaxf(h4.x, 0.f); s0 += h * W2[(q * 4 + 0) * 2 + 0]; s1 += h * W2[(q * 4 + 0) * 2 + 1];
        h = fmaxf(h4.y, 0.f); s0 += h * W2[(q * 4 + 1) * 2 + 0]; s1 += h * W2[(q * 4 + 1) * 2 + 1];
        h = fmaxf(h4.z, 0.f); s0 += h * W2[(q * 4 + 2) * 2 + 0]; s1 += h * W2[(q * 4 + 2) * 2 + 1];
        h = fmaxf(h4.w, 0.f); s0 += h * W2[(q * 4 + 3) * 2 + 0]; s1 += h * W2[(q * 4 + 3) * 2 + 1];
    }
    float di2 = dinv[i] * dinv[i];
    hw2[i * 2 + 0] = s0;
    hw2[i * 2 + 1] = s1;
    acc2[i * 2 + 0] = b2[0] + di2 * s0;
    acc2[i * 2 + 1] = b2[1] + di2 * s1;
}

__global__ __launch_bounds__(256) void k_edge2(const int* __restrict__ ei,
                                               const float* __restrict__ dinv,
                                               const float* __restrict__ hw2,
                                               float* __restrict__ acc2, int e) {
    int edge = blockIdx.x * blockDim.x + threadIdx.x;
    if (edge >= e) return;
    int s = ei[edge];
    int d = ei[e + edge];
    float nrm = dinv[s] * dinv[d];
    atomicAdd(&acc2[d * 2 + 0], nrm * hw2[s * 2 + 0]);
    atomicAdd(&acc2[d * 2 + 1], nrm * hw2[s * 2 + 1]);
}

__global__ __launch_bounds__(256) void k_logsoftmax(const float* __restrict__ acc2,
                                                    float* __restrict__ out, int n) {
    int i = blockIdx.x * blockDim.x + threadIdx.x;
    if (i >= n) return;
    float z0 = acc2[i * 2 + 0];
    float z1 = acc2[i * 2 + 1];
    float m = fmaxf(z0, z1);
    float lse = m + logf(expf(z0 - m) + expf(z1 - m));
    out[i * 2 + 0] = z0 - lse;
    out[i * 2 + 1] = z1 - lse;
}

// ---------------------------------------------------------------------------
extern "C" void kernel_launch(void* const* d_in, const int* in_sizes, int n_in,
                              void* d_out, int out_size, void* d_ws, size_t ws_size,
                              hipStream_t stream) {
    const float* x  = (const float*)d_in[0];
    const int*   ei = (const int*)d_in[1];
    const float* W1 = (const float*)d_in[2];
    const float* b1 = (const float*)d_in[3];
    const float* W2 = (const float*)d_in[4];
    const float* b2 = (const float*)d_in[5];

    const int N = in_sizes[0] / IN_DIM;   // 50000
    const int E = in_sizes[1] / 2;        // 800000

    // workspace layout (floats): dinv[N] | hW[N*128] | acc1[N*128] | hw2[2N] | acc2[2N]
    float* dinv = (float*)d_ws;
    float* hW   = dinv + N;
    float* acc1 = hW + (size_t)N * HID;
    float* hw2  = acc1 + (size_t)N * HID;
    float* acc2 = hw2 + (size_t)N * 2;

    float* out = (float*)d_out;

    // 1) symmetric GCN normalization
    k_init_deg<<<(N + 255) / 256, 256, 0, stream>>>(dinv, N);
    k_deg_accum<<<(E + 255) / 256, 256, 0, stream>>>(ei + E, dinv, E);
    k_dinv<<<(N + 255) / 256, 256, 0, stream>>>(dinv, N);

    // 2) hW = x @ W1 via f32 WMMA
    k_gemm1<<<N / 16, 128, 0, stream>>>(x, W1, hW, N);

    // 3) layer-1 aggregation (self-loop fused into init) + edge scatter
    k_init_acc1<<<((N * HID) + 255) / 256, 256, 0, stream>>>(hW, dinv, b1, acc1, N);
    k_edge1<<<E, 128, 0, stream>>>(ei, dinv, hW, acc1, E);

    // 4) layer 2: relu-on-read dot with W2, self-loop init of acc2
    k_layer2<<<(N + 127) / 128, 128, 0, stream>>>(acc1, dinv, W2, b2, hw2, acc2, N);
    k_edge2<<<(E + 255) / 256, 256, 0, stream>>>(ei, dinv, hw2, acc2, E);

    // 5) log-softmax over the 2 classes
    k_logsoftmax<<<(N + 255) / 256, 256, 0, stream>>>(acc2, out, N);
}